// GNN_MTA_62225486184607
// MI455X (gfx1250) — compile-verified
//
#include <hip/hip_runtime.h>

#define N_NODES   50000
#define N_EDGES   800000
#define EMB       100
#define N_LAYERS  5
#define N_GRAPHS  512
#define KMAX      200          // largest K at any GEMM call site
#define WAVES_PER_BLOCK 8

typedef __attribute__((ext_vector_type(2))) float v2f;
typedef __attribute__((ext_vector_type(8))) float v8f;

// ---------------------------------------------------------------------------
// agg = (1 + eps[layer]) * h          (elementwise, grid-stride)
// ---------------------------------------------------------------------------
__global__ void scale_init_kernel(const float* __restrict__ h,
                                  float* __restrict__ agg,
                                  const float* __restrict__ eps,
                                  int layer, int n)
{
    float s = 1.0f + eps[layer];
    int i = blockIdx.x * blockDim.x + threadIdx.x;
    int stride = gridDim.x * blockDim.x;
    for (; i < n; i += stride)
        agg[i] = s * h[i];
}

// ---------------------------------------------------------------------------
// agg[dst] += h[src] over all edges. One wave per edge; 100 floats / 32 lanes
// -> 3 full vectors + 4-lane tail. Node features (20 MB) are L2-resident.
// ---------------------------------------------------------------------------
__global__ void gin_aggregate_kernel(const float* __restrict__ h,
                                     const int* __restrict__ src,
                                     const int* __restrict__ dst,
                                     float* __restrict__ agg)
{
    int wid   = (blockIdx.x * blockDim.x + threadIdx.x) >> 5;
    int lane  = threadIdx.x & 31;
    int nwave = (gridDim.x * blockDim.x) >> 5;

    for (int e = wid; e < N_EDGES; e += nwave) {
        if (e + nwave < N_EDGES) {
            __builtin_prefetch(&src[e + nwave], 0, 0);
            __builtin_prefetch(&dst[e + nwave], 0, 0);
        }
        int s = src[e];
        int d = dst[e];
        const float* hs = h   + (size_t)s * EMB;
        float*       ad = agg + (size_t)d * EMB;
        atomicAdd(&ad[lane],      hs[lane]);
        atomicAdd(&ad[lane + 32], hs[lane + 32]);
        atomicAdd(&ad[lane + 64], hs[lane + 64]);
        if (lane < EMB - 96)
            atomicAdd(&ad[lane + 96], hs[lane + 96]);
    }
}

// ---------------------------------------------------------------------------
// C[M x N] = act(A[M x K] @ B[K x N] + bias), f32 WMMA 16x16x4.
//
// Grid: (ceil(tilesM/8), tilesN). All 8 waves of a block share one 16-column
// B panel, staged zero-padded in LDS (K*16 floats <= 12.8 KB). Fill uses
// clamped addresses + select (no predicated loads). Out-of-range M tiles are
// clamped for loads and guarded only at the store, so every wave reaches the
// barrier and EXEC is all-1s at each WMMA.
//
// Requires M % 16 == 0, K % 4 == 0, K <= KMAX (true at all call sites).
// ---------------------------------------------------------------------------
__global__ void __launch_bounds__(32 * WAVES_PER_BLOCK)
wmma_gemm_kernel(const float* __restrict__ A,
                 const float* __restrict__ B,
                 const float* __restrict__ bias,
                 float* __restrict__ C,
                 int M, int N, int K, int relu)
{
    __shared__ float Bs[KMAX * 16];

    int tid    = threadIdx.x;
    int lane   = tid & 31;
    int wv     = tid >> 5;                     // wave index within block
    int tilesM = M >> 4;
    int tn     = blockIdx.y;
    int tm     = blockIdx.x * WAVES_PER_BLOCK + wv;
    bool tmOK  = tm < tilesM;
    int  tmc   = tmOK ? tm : tilesM - 1;       // clamp: loads stay in-bounds

    // ---- cooperative fill of the zero-padded B panel ----
    for (int i = tid; i < K * 16; i += 32 * WAVES_PER_BLOCK) {
        int k   = i >> 4;
        int c   = i & 15;
        int col = (tn << 4) + c;
        int colC = col < N ? col : N - 1;      // clamped -> load never OOB
        float v  = B[(size_t)k * N + colC];
        Bs[i] = (col < N) ? v : 0.0f;
    }
    __syncthreads();

    int r15  = lane & 15;
    int hi   = lane >> 4;        // 0 = low lane half, 1 = high lane half
    int koff = hi << 1;          // K sub-offset owned by this lane half

    int row = (tmc << 4) + r15;
    int col = (tn << 4) + r15;

    const float* Arow = A + (size_t)row * K;

    v8f acc = {};

#if __has_builtin(__builtin_amdgcn_wmma_f32_16x16x4_f32)
    for (int k = 0; k < K; k += 4) {
        v2f a, b;
        a.x = Arow[k + koff];                  // contiguous pair -> b64 load
        a.y = Arow[k + koff + 1];
        b.x = Bs[(k + koff)     * 16 + r15];   // LDS, conflict-free
        b.y = Bs[(k + koff + 1) * 16 + r15];
        acc = __builtin_amdgcn_wmma_f32_16x16x4_f32(
            false, a, false, b, (short)0, acc, false, false);
    }
#else
    for (int r = 0; r < 8; ++r) {
        int orow = (tmc << 4) + r + (hi << 3);
        float s = 0.0f;
        for (int k = 0; k < K; ++k)
            s += A[(size_t)orow * K + k] * Bs[k * 16 + r15];
        acc[r] = s;
    }
#endif

    if (tmOK && col < N) {
        float bv = bias[col];
#pragma unroll
        for (int r = 0; r < 8; ++r) {
            int orow = (tm << 4) + r + (hi << 3);
            float v = acc[r] + bv;
            if (relu) v = fmaxf(v, 0.0f);
            C[(size_t)orow * N + col] = v;
        }
    }
}

// ---------------------------------------------------------------------------
// zero a float buffer (harness poisons workspace; pooling uses atomics)
// ---------------------------------------------------------------------------
__global__ void zero_kernel(float* __restrict__ p, int n)
{
    int i = blockIdx.x * blockDim.x + threadIdx.x;
    int stride = gridDim.x * blockDim.x;
    for (; i < n; i += stride) p[i] = 0.0f;
}

// ---------------------------------------------------------------------------
// fused global_add_pool for both heads: g{1,2}[batch[n]] += hn{1,2}[n]
// ---------------------------------------------------------------------------
__global__ void pool2_kernel(const float* __restrict__ hn1,
                             const float* __restrict__ hn2,
                             const int* __restrict__ batch,
                             float* __restrict__ g1,
                             float* __restrict__ g2)
{
    int wid   = (blockIdx.x * blockDim.x + threadIdx.x) >> 5;
    int lane  = threadIdx.x & 31;
    int nwave = (gridDim.x * blockDim.x) >> 5;

    for (int n = wid; n < N_NODES; n += nwave) {
        int gidx = batch[n];
        const float* a = hn1 + (size_t)n * EMB;
        const float* b = hn2 + (size_t)n * EMB;
        float* o1 = g1 + (size_t)gidx * EMB;
        float* o2 = g2 + (size_t)gidx * EMB;
        atomicAdd(&o1[lane],      a[lane]);
        atomicAdd(&o1[lane + 32], a[lane + 32]);
        atomicAdd(&o1[lane + 64], a[lane + 64]);
        atomicAdd(&o2[lane],      b[lane]);
        atomicAdd(&o2[lane + 32], b[lane + 32]);
        atomicAdd(&o2[lane + 64], b[lane + 64]);
        if (lane < EMB - 96) {
            atomicAdd(&o1[lane + 96], a[lane + 96]);
            atomicAdd(&o2[lane + 96], b[lane + 96]);
        }
    }
}

static inline dim3 gemm_grid(int M, int N)
{
    int tilesM = M >> 4;
    int tilesN = (N + 15) >> 4;
    return dim3((tilesM + WAVES_PER_BLOCK - 1) / WAVES_PER_BLOCK, tilesN, 1);
}

extern "C" void kernel_launch(void* const* d_in, const int* in_sizes, int n_in,
                              void* d_out, int out_size, void* d_ws, size_t ws_size,
                              hipStream_t stream)
{
    const float* x     = (const float*)d_in[0];
    const int*   ei    = (const int*)  d_in[1];
    const int*   batch = (const int*)  d_in[2];
    const float* W1    = (const float*)d_in[3];
    const float* b1    = (const float*)d_in[4];
    const float* W2    = (const float*)d_in[5];
    const float* b2    = (const float*)d_in[6];
    const float* eps   = (const float*)d_in[7];
    const float* Wh1   = (const float*)d_in[8];
    const float* bh1   = (const float*)d_in[9];
    const float* Wh2   = (const float*)d_in[10];
    const float* bh2   = (const float*)d_in[11];
    const float* Wp1   = (const float*)d_in[12];
    const float* bp1   = (const float*)d_in[13];
    const float* Wp2   = (const float*)d_in[14];
    const float* bp2   = (const float*)d_in[15];
    float* out = (float*)d_out;

    // workspace layout (floats): h | agg | z1 | g   (~80.4 MB total)
    float* ws  = (float*)d_ws;
    float* h   = ws;                                    // 50000*100
    float* agg = h   + (size_t)N_NODES * EMB;           // 50000*100
    float* z1  = agg + (size_t)N_NODES * EMB;           // 50000*200
    float* g   = z1  + (size_t)N_NODES * 2 * EMB;       // 2*512*100
    float* hn1 = agg;   // head-1 node embeddings reuse agg after layer loop
    float* hn2 = z1;    // head-2 node embeddings reuse z1

    const int* src = ei;             // edge_index[0]
    const int* dst = ei + N_EDGES;   // edge_index[1]

    const int elems = N_NODES * EMB;
    const int blk   = 32 * WAVES_PER_BLOCK;

    // h = x
    hipMemcpyAsync(h, x, sizeof(float) * (size_t)elems,
                   hipMemcpyDeviceToDevice, stream);

    // -------- GIN layers --------
    for (int l = 0; l < N_LAYERS; ++l) {
        scale_init_kernel<<<4096, blk, 0, stream>>>(h, agg, eps, l, elems);
        gin_aggregate_kernel<<<2048, blk, 0, stream>>>(h, src, dst, agg);

        // z1 = relu(agg @ W1[l] + b1[l])       [50000 x 200]
        wmma_gemm_kernel<<<gemm_grid(N_NODES, 2 * EMB), blk, 0, stream>>>(
            agg, W1 + (size_t)l * EMB * 2 * EMB, b1 + (size_t)l * 2 * EMB,
            z1, N_NODES, 2 * EMB, EMB, 1);

        // h = (relu?)(z1 @ W2[l] + b2[l])      [50000 x 100]
        wmma_gemm_kernel<<<gemm_grid(N_NODES, EMB), blk, 0, stream>>>(
            z1, W2 + (size_t)l * 2 * EMB * EMB, b2 + (size_t)l * EMB,
            h, N_NODES, EMB, 2 * EMB, (l < N_LAYERS - 1) ? 1 : 0);
    }

    // -------- MTA heads --------
    wmma_gemm_kernel<<<gemm_grid(N_NODES, EMB), blk, 0, stream>>>(
        h, Wh1, bh1, hn1, N_NODES, EMB, EMB, 0);
    wmma_gemm_kernel<<<gemm_grid(N_NODES, EMB), blk, 0, stream>>>(
        h, Wh2, bh2, hn2, N_NODES, EMB, EMB, 0);

    // -------- global add pool --------
    float* g1 = g;
    float* g2 = g + (size_t)N_GRAPHS * EMB;
    zero_kernel<<<128, blk, 0, stream>>>(g, 2 * N_GRAPHS * EMB);
    pool2_kernel<<<1024, blk, 0, stream>>>(hn1, hn2, batch, g1, g2);

    // -------- graph prediction heads (straight into d_out) --------
    wmma_gemm_kernel<<<gemm_grid(N_GRAPHS, EMB), blk, 0, stream>>>(
        g1, Wp1, bp1, out, N_GRAPHS, EMB, EMB, 0);
    wmma_gemm_kernel<<<gemm_grid(N_GRAPHS, EMB), blk, 0, stream>>>(
        g2, Wp2, bp2, out + (size_t)N_GRAPHS * EMB, N_GRAPHS, EMB, EMB, 0);
}